// QuantumCircuit_2027224564654
// MI455X (gfx1250) — compile-verified
//
#include <hip/hip_runtime.h>
#include <cstddef>
#include <cstdint>

#define NUM_Q   2048
#define NUM_LAY 8
#define BATCH_M 4096
#define GN      2048   // N (columns of entanglement matrix)
#define GK      2048   // K (rows of entanglement matrix)

typedef __attribute__((ext_vector_type(2))) float v2f;
typedef __attribute__((ext_vector_type(8))) float v8f;

#define BM 128
#define BN 128
#define BK 16
#define LDAP 20   // A tile stride (floats): 80B rows, 16B aligned, conflict-free
#define LDBP 18   // B tile stride (floats): 72B rows, 8B aligned, conflict-free

// gfx1250 async global->LDS DMA (ASYNCcnt-tracked), with staged-copy fallback.
#if defined(__HIP_DEVICE_COMPILE__) &&                                         \
    __has_builtin(__builtin_amdgcn_global_load_async_to_lds_b128) &&           \
    __has_builtin(__builtin_amdgcn_global_load_async_to_lds_b32)
#define HAVE_ASYNC_LDS 1

typedef int v4i __attribute__((__vector_size__(16)));
typedef __attribute__((address_space(1))) v4i* glb_v4i_p;
typedef __attribute__((address_space(3))) v4i* lds_v4i_p;
typedef __attribute__((address_space(1))) int* glb_i32_p;
typedef __attribute__((address_space(3))) int* lds_i32_p;

__device__ __forceinline__ void async_copy_b128(const float* gp, float* lp) {
  __builtin_amdgcn_global_load_async_to_lds_b128((glb_v4i_p)gp, (lds_v4i_p)lp,
                                                 0, 0);
}
__device__ __forceinline__ void async_copy_b32(const float* gp, float* lp) {
  __builtin_amdgcn_global_load_async_to_lds_b32((glb_i32_p)gp, (lds_i32_p)lp,
                                                0, 0);
}
#else
#define HAVE_ASYNC_LDS 0
#endif

__device__ __forceinline__ void wait_async_zero() {
#if defined(__HIP_DEVICE_COMPILE__)
#if __has_builtin(__builtin_amdgcn_s_wait_asynccnt)
  __builtin_amdgcn_s_wait_asynccnt(0);
#else
  asm volatile("s_wait_asynccnt 0x0" ::: "memory");
#endif
#endif
}

// ---------------------------------------------------------------------------
// Per-layer affine closed form: scale = prod_q cos(r0_q),
// bias = sum_q sin(r1_q) * prod_{j>q} cos(r0_j).  One wave per layer.
// ---------------------------------------------------------------------------
__global__ __launch_bounds__(32)
void qc_layer_scalars(const float* __restrict__ rot, float* __restrict__ sb) {
  __shared__ float Ps[32];
  __shared__ float Bsh[32];
  const int l = blockIdx.x;
  const int t = threadIdx.x;
  const float* base = rot + (size_t)l * NUM_Q * 3;
  float P = 1.0f, B = 0.0f;
  const int q0 = t * (NUM_Q / 32);
  for (int i = 0; i < NUM_Q / 32; ++i) {
    const int q = q0 + i;
    const float c = cosf(base[q * 3 + 0]);
    const float s = sinf(base[q * 3 + 1]);
    P *= c;
    B = B * c + s;
  }
  Ps[t] = P;
  Bsh[t] = B;
  __syncthreads();
  if (t == 0) {
    float Pt = 1.0f, Bt = 0.0f;
    for (int i = 0; i < 32; ++i) {
      Bt = Bt * Ps[i] + Bsh[i];
      Pt *= Ps[i];
    }
    sb[2 * l]     = Pt;
    sb[2 * l + 1] = Bt;
  }
}

// ---------------------------------------------------------------------------
// colsum[l][n] = sum_k W_l[k][n]  (lets the GEMM epilogue absorb the affine:
// (A*s + b) @ W = s*(A@W) + b*colsum(W), so tile staging is a pure copy)
// ---------------------------------------------------------------------------
__global__ __launch_bounds__(256)
void qc_colsum(const float* __restrict__ ent, float* __restrict__ cs) {
  const int l = blockIdx.y;
  const int n = blockIdx.x * 256 + threadIdx.x;
  const float* W = ent + (size_t)l * GK * GN;
  float s = 0.0f;
#pragma unroll 8
  for (int k = 0; k < GK; ++k) s += W[(size_t)k * GN + n];
  cs[l * GN + n] = s;
}

// ---------------------------------------------------------------------------
// C = scale*(A @ W) + bias*colsum(W) via V_WMMA_F32_16X16X4_F32.
// 128x128 block tile, BK=16, double-buffered LDS fed by async global->LDS DMA.
// 8 waves; each wave owns a 64x32 sub-tile = 4x2 WMMA tiles (64 acc VGPRs).
// ---------------------------------------------------------------------------
__global__ __launch_bounds__(256)
void qc_gemm(const float* __restrict__ A,
             const float* __restrict__ W,
             float* __restrict__ C,
             const float* __restrict__ sb,
             const float* __restrict__ cs,   // this layer's colsum [GN]
             int layer) {
  __shared__ alignas(16) float As[2][BM * LDAP];   // [m][k]
  __shared__ alignas(16) float Bs[2][BN * LDBP];   // transposed: [n][k]

  const int tid  = threadIdx.x;
  const int lane = tid & 31;
  const int wave = tid >> 5;
  const int wm   = wave >> 2;          // 0..1 -> m offset 64*wm
  const int wn   = wave & 3;           // 0..3 -> n offset 32*wn
  const int fm   = lane & 15;          // row (A) / col (B) within 16
  const int kb   = (lane >> 4) << 1;   // K sub-offset: 0 (lanes 0-15) or 2

  const int m0 = blockIdx.y * BM;
  const int n0 = blockIdx.x * BN;

  v8f acc[4][2] = {};

  auto load_tiles = [&](int k0, int buf) {
#if HAVE_ASYNC_LDS
    // A tile: 128x16 floats, 2 x b128 per thread, straight DMA into LDS.
#pragma unroll
    for (int p = 0; p < 2; ++p) {
      const int r  = (tid >> 2) + p * 64;     // 0..127
      const int c4 = (tid & 3) * 4;           // 0,4,8,12
      async_copy_b128(A + (size_t)(m0 + r) * GK + k0 + c4,
                      &As[buf][r * LDAP + c4]);
    }
    // B tile: 16x128 floats, transposed via per-element b32 DMA scatter.
#pragma unroll
    for (int p = 0; p < 8; ++p) {
      const int e  = p * 256 + tid;           // 0..2047
      const int kr = e >> 7;                  // 0..15
      const int n  = e & 127;                 // 0..127
      async_copy_b32(W + (size_t)(k0 + kr) * GN + n0 + n,
                     &Bs[buf][n * LDBP + kr]);
    }
#else
    // Fallback: staged copy through VGPRs.
#pragma unroll
    for (int p = 0; p < 2; ++p) {
      const int r  = (tid >> 2) + p * 64;
      const int c4 = (tid & 3) * 4;
      const float4 v = *(const float4*)(A + (size_t)(m0 + r) * GK + k0 + c4);
      *(float4*)(&As[buf][r * LDAP + c4]) = v;
    }
#pragma unroll
    for (int p = 0; p < 8; ++p) {
      const int e  = p * 256 + tid;
      const int kr = e >> 7;
      const int n  = e & 127;
      Bs[buf][n * LDBP + kr] = W[(size_t)(k0 + kr) * GN + n0 + n];
    }
#endif
  };

  load_tiles(0, 0);
  int buf = 0;

  for (int k0 = 0; k0 < GK; k0 += BK) {
    wait_async_zero();     // my DMA into `buf` has landed
    __syncthreads();       // everyone's DMA landed; prev reads of buf^1 done
    if (k0 + BK < GK) load_tiles(k0 + BK, buf ^ 1);  // overlap with compute

#pragma unroll
    for (int kk = 0; kk < BK; kk += 4) {
      v2f bfrag[2];
#pragma unroll
      for (int j = 0; j < 2; ++j) {
        const int n = wn * 32 + j * 16 + fm;
        bfrag[j] = *(const v2f*)(&Bs[buf][n * LDBP + kk + kb]);
      }
#pragma unroll
      for (int i = 0; i < 4; ++i) {
        const int m = wm * 64 + i * 16 + fm;
        const v2f afrag = *(const v2f*)(&As[buf][m * LDAP + kk + kb]);
#pragma unroll
        for (int j = 0; j < 2; ++j) {
          acc[i][j] = __builtin_amdgcn_wmma_f32_16x16x4_f32(
              false, afrag, false, bfrag[j], (short)0, acc[i][j],
              false, false);
        }
      }
    }
    buf ^= 1;
  }

  // Epilogue: out = scale*acc + bias*colsum[n].
  const float scale = sb[2 * layer];
  const float bias  = sb[2 * layer + 1];
  float cb[2];
#pragma unroll
  for (int j = 0; j < 2; ++j)
    cb[j] = bias * cs[n0 + wn * 32 + j * 16 + fm];

#pragma unroll
  for (int i = 0; i < 4; ++i) {
#pragma unroll
    for (int j = 0; j < 2; ++j) {
#pragma unroll
      for (int v = 0; v < 8; ++v) {
        const int gm = m0 + wm * 64 + i * 16 + v + ((lane >> 4) << 3);
        const int gn = n0 + wn * 32 + j * 16 + fm;
        C[(size_t)gm * GN + gn] = scale * acc[i][j][v] + cb[j];
      }
    }
  }
}

// ---------------------------------------------------------------------------
extern "C" void kernel_launch(void* const* d_in, const int* in_sizes, int n_in,
                              void* d_out, int out_size, void* d_ws, size_t ws_size,
                              hipStream_t stream) {
  const float* x   = (const float*)d_in[0];  // [4096, 2048]
  const float* rot = (const float*)d_in[1];  // [8, 2048, 3]
  const float* ent = (const float*)d_in[2];  // [8, 2048, 2048]
  float* out = (float*)d_out;                // [4096, 2048]

  float* sb  = (float*)d_ws;                                   // 16 floats
  float* cs  = (float*)((char*)d_ws + 256);                    // 8*2048 floats
  float* buf = (float*)((char*)d_ws + 256 + NUM_LAY * GN * 4); // ping buffer

  qc_layer_scalars<<<NUM_LAY, 32, 0, stream>>>(rot, sb);
  qc_colsum<<<dim3(GN / 256, NUM_LAY), 256, 0, stream>>>(ent, cs);

  dim3 grid(GN / BN, BATCH_M / BM);   // (16, 32)
  const float* in = x;
  for (int l = 0; l < NUM_LAY; ++l) {
    float* ob = (l & 1) ? out : buf;  // layer 7 (odd) lands in d_out
    qc_gemm<<<grid, 256, 0, stream>>>(in, ent + (size_t)l * GK * GN,
                                      ob, sb, cs + l * GN, l);
    in = ob;
  }
}